// GCNClassifier_10282151707007
// MI455X (gfx1250) — compile-verified
//
#include <hip/hip_runtime.h>
#include <hip/hip_bf16.h>

// GCN classifier for MI455X (gfx1250, wave32).
// Key optimization: the (B,S,S,E) intermediate adjacency update is NEVER
// materialized. Layer 1 only consumes sum_e(adj1), which collapses to a
// dot of the original adj with column-sums of Wh_e plus rank-1 terms.
// => exactly one 118MB pass over adj (bandwidth bound, ~5us at 23.3TB/s),
// streamed with non-temporal b64 loads; all dense math runs on
// V_WMMA_F32_16X16X4_F32 (f32-exact matrix core).

typedef __attribute__((ext_vector_type(2))) float v2f;
typedef __attribute__((ext_vector_type(8))) float v8f;

#define Smax 384
#define Dm   300
#define Em   50
#define HDm  150
#define Bm   4

static __device__ __forceinline__ v8f wmma4(v2f a, v2f b, v8f c) {
  // D = A(16x4,f32) * B(4x16,f32) + C(16x16,f32)
  return __builtin_amdgcn_wmma_f32_16x16x4_f32(false, a, false, b, (short)0, c,
                                               false, false);
}

// ---- K0: fold Wh/bh into column-sum vectors --------------------------------
__global__ void k0_sums(const float* __restrict__ Wh, const float* __restrict__ bh,
                        float* __restrict__ we, float* __restrict__ wh1,
                        float* __restrict__ wh2, float* __restrict__ bhs) {
  const int t = threadIdx.x;
  const int LD = 2 * HDm + Em; // 350
  if (t < Em) {
    float s = 0.f;
    for (int e = 0; e < Em; ++e) s += Wh[e * LD + t];
    we[t] = s;
  }
  if (t < HDm) {
    float s1 = 0.f, s2 = 0.f;
    for (int e = 0; e < Em; ++e) {
      s1 += Wh[e * LD + Em + t];
      s2 += Wh[e * LD + Em + HDm + t];
    }
    wh1[t] = s1;
    wh2[t] = s2;
  }
  if (t == 0) {
    float s = 0.f;
    for (int e = 0; e < Em; ++e) s += bh[e];
    *bhs = s;
  }
}

// ---- K1: single NT-streamed pass over adj (118MB) -> A2_0 and A2w ----------
__global__ void k1_adjreduce(const float* __restrict__ adj, const float* __restrict__ we,
                             float* __restrict__ A20, float* __restrict__ A2w) {
  __shared__ float wc[Em];
  if (threadIdx.x < Em) wc[threadIdx.x] = we[threadIdx.x];
  __syncthreads();
  size_t t = (size_t)blockIdx.x * blockDim.x + threadIdx.x;
  if (t >= (size_t)Bm * Smax * Smax) return;
  // row stride is 200B -> 8B-aligned; stream as 25 x b64 non-temporal loads
  const v2f* p2 = (const v2f*)(adj + t * Em);
  float s = 0.f, w = 0.f;
#pragma unroll
  for (int j = 0; j < Em / 2; ++j) {
    const v2f v = __builtin_nontemporal_load(p2 + j);
    s += v.x + v.y;
    w += v.x * wc[2 * j] + v.y * wc[2 * j + 1];
  }
  A20[t] = s * 0.02f;  // 1/E
  A2w[t] = w;          // finished into A2_1 later
}

// ---- K2: Y0 = A2_0 @ X + X   (per-batch 384x384 @ 384x300) -----------------
__global__ void k2_gemm_y0(const float* __restrict__ A2, const float* __restrict__ X,
                           float* __restrict__ Y0) {
  const int wave = (blockIdx.x * blockDim.x + threadIdx.x) >> 5;
  const int lane = threadIdx.x & 31;
  int nt = wave % 19, tmp = wave / 19, mt = tmp % 24, b = tmp / 24;
  const int l16 = lane & 15, half = lane >> 4;
  const int m0 = mt * 16, n = nt * 16 + l16;
  const bool nOK = n < Dm;
  const int nc = nOK ? n : 0;
  const float* Ab = A2 + (size_t)b * Smax * Smax;
  const float* Xb = X + (size_t)b * Smax * Dm;
  const float* Arow = Ab + (size_t)(m0 + l16) * Smax;
  v8f c = {};
  for (int k0 = 0; k0 < Smax; k0 += 4) {
    const int ka = k0 + 2 * half;
    v2f a, bb;
    a.x = Arow[ka];
    a.y = Arow[ka + 1];
    float bv0 = Xb[(size_t)ka * Dm + nc];
    float bv1 = Xb[(size_t)(ka + 1) * Dm + nc];
    bb.x = nOK ? bv0 : 0.f;
    bb.y = nOK ? bv1 : 0.f;
    c = wmma4(a, bb, c);
  }
  if (nOK) {
#pragma unroll
    for (int r = 0; r < 8; ++r) {
      const int m = m0 + r + 8 * half;
      Y0[((size_t)b * Smax + m) * Dm + n] = c[r] + Xb[(size_t)m * Dm + n];
    }
  }
}

// ---- K3: h0 = relu(Y0 @ W0^T + 2*b0)   (1536x300 @ 300x150) ----------------
__global__ void k3_gemm_h0(const float* __restrict__ Y0, const float* __restrict__ W0,
                           const float* __restrict__ b0, float* __restrict__ h0) {
  const int wave = (blockIdx.x * blockDim.x + threadIdx.x) >> 5;
  const int lane = threadIdx.x & 31;
  int nt = wave % 10, mt = wave / 10;
  const int l16 = lane & 15, half = lane >> 4;
  const int m0 = mt * 16, n = nt * 16 + l16;
  const bool nOK = n < HDm;
  const int nc = nOK ? n : 0;
  const float* Arow = Y0 + (size_t)(m0 + l16) * Dm;
  const float* Wn = W0 + (size_t)nc * Dm;  // B[k][n] = W0[n][k]
  v8f c = {};
  for (int k0 = 0; k0 < Dm; k0 += 4) {
    const int ka = k0 + 2 * half;
    v2f a, bb;
    a.x = Arow[ka];
    a.y = Arow[ka + 1];
    float w0v = Wn[ka], w1v = Wn[ka + 1];
    bb.x = nOK ? w0v : 0.f;
    bb.y = nOK ? w1v : 0.f;
    c = wmma4(a, bb, c);
  }
  if (nOK) {
    const float bias = 2.f * b0[n];
#pragma unroll
    for (int r = 0; r < 8; ++r) {
      const int m = m0 + r + 8 * half;
      const float v = c[r] + bias;
      h0[(size_t)m * HDm + n] = v > 0.f ? v : 0.f;
    }
  }
}

// ---- K4: rank-1 terms s1[b,k] = h0·wh1_sum, s2[b,i] = h0·wh2_sum -----------
__global__ void k4_s12(const float* __restrict__ h0, const float* __restrict__ wh1,
                       const float* __restrict__ wh2, float* __restrict__ s1,
                       float* __restrict__ s2) {
  const int r = blockIdx.x * blockDim.x + threadIdx.x;
  if (r >= Bm * Smax) return;
  const float* hr = h0 + (size_t)r * HDm;
  float a = 0.f, b = 0.f;
  for (int d = 0; d < HDm; ++d) {
    const float v = hr[d];
    a += v * wh1[d];
    b += v * wh2[d];
  }
  s1[r] = a;
  s2[r] = b;
}

// ---- K5: A2_1 = (A2w + s1[b,k] + s2[b,i] + bh_sum) / E  (in place) ---------
__global__ void k5_a21(float* __restrict__ A2w, const float* __restrict__ s1,
                       const float* __restrict__ s2, const float* __restrict__ bhs) {
  size_t t = (size_t)blockIdx.x * blockDim.x + threadIdx.x;
  if (t >= (size_t)Bm * Smax * Smax) return;
  const int k = (int)(t % Smax);
  const size_t u = t / Smax;
  const int i = (int)(u % Smax);
  const int b = (int)(u / Smax);
  A2w[t] = (A2w[t] + s1[b * Smax + k] + s2[b * Smax + i] + *bhs) * 0.02f;
}

// ---- K6: Y1 = A2_1 @ [X | h0] + [X | h0]  (384x384 @ 384x450) --------------
__global__ void k6_gemm_y1(const float* __restrict__ A2, const float* __restrict__ X,
                           const float* __restrict__ h0, float* __restrict__ Y1) {
  const int wave = (blockIdx.x * blockDim.x + threadIdx.x) >> 5;
  const int lane = threadIdx.x & 31;
  int nt = wave % 29, tmp = wave / 29, mt = tmp % 24, b = tmp / 24;
  const int l16 = lane & 15, half = lane >> 4;
  const int m0 = mt * 16, n = nt * 16 + l16;
  const float* Ab = A2 + (size_t)b * Smax * Smax;
  const float* Xb = X + (size_t)b * Smax * Dm;
  const float* Hb = h0 + (size_t)b * Smax * HDm;
  const float* Arow = Ab + (size_t)(m0 + l16) * Smax;
  const bool inX = n < Dm;
  const bool inH = (n >= Dm) && (n < Dm + HDm);
  const int nx = inX ? n : 0;
  const int nh = inH ? (n - Dm) : 0;
  v8f c = {};
  for (int k0 = 0; k0 < Smax; k0 += 4) {
    const int ka = k0 + 2 * half;
    v2f a, bb;
    a.x = Arow[ka];
    a.y = Arow[ka + 1];
    float xv0 = Xb[(size_t)ka * Dm + nx], xv1 = Xb[(size_t)(ka + 1) * Dm + nx];
    float hv0 = Hb[(size_t)ka * HDm + nh], hv1 = Hb[(size_t)(ka + 1) * HDm + nh];
    bb.x = inX ? xv0 : (inH ? hv0 : 0.f);
    bb.y = inX ? xv1 : (inH ? hv1 : 0.f);
    c = wmma4(a, bb, c);
  }
  if (n < Dm + HDm) {
#pragma unroll
    for (int r = 0; r < 8; ++r) {
      const int m = m0 + r + 8 * half;
      const float add = inX ? Xb[(size_t)m * Dm + nx] : Hb[(size_t)m * HDm + nh];
      Y1[((size_t)b * Smax + m) * (Dm + HDm) + n] = c[r] + add;
    }
  }
}

// ---- K7: h1 = relu(Y1 @ W1^T + 2*b1)   (1536x450 @ 450x150, K tail) --------
__global__ void k7_gemm_h1(const float* __restrict__ Y1, const float* __restrict__ W1,
                           const float* __restrict__ b1, float* __restrict__ h1) {
  const int wave = (blockIdx.x * blockDim.x + threadIdx.x) >> 5;
  const int lane = threadIdx.x & 31;
  int nt = wave % 10, mt = wave / 10;
  const int l16 = lane & 15, half = lane >> 4;
  const int m0 = mt * 16, n = nt * 16 + l16;
  const bool nOK = n < HDm;
  const int nc = nOK ? n : 0;
  const int K = Dm + HDm;  // 450 (not a multiple of 4 -> tail guards)
  const float* Arow = Y1 + (size_t)(m0 + l16) * K;
  const float* Wn = W1 + (size_t)nc * K;
  v8f c = {};
  for (int k0 = 0; k0 < 450; k0 += 4) {
    const int ka = k0 + 2 * half;
    const bool kok0 = ka < K, kok1 = (ka + 1) < K;
    const int kc0 = kok0 ? ka : 0, kc1 = kok1 ? (ka + 1) : 0;
    v2f a, bb;
    float a0 = Arow[kc0], a1 = Arow[kc1];
    a.x = kok0 ? a0 : 0.f;
    a.y = kok1 ? a1 : 0.f;
    float w0v = Wn[kc0], w1v = Wn[kc1];
    bb.x = (nOK && kok0) ? w0v : 0.f;
    bb.y = (nOK && kok1) ? w1v : 0.f;
    c = wmma4(a, bb, c);
  }
  if (nOK) {
    const float bias = 2.f * b1[n];
#pragma unroll
    for (int r = 0; r < 8; ++r) {
      const int m = m0 + r + 8 * half;
      const float v = c[r] + bias;
      h1[(size_t)m * HDm + n] = v > 0.f ? v : 0.f;
    }
  }
}

// ---- K8: out = ([h0|h1] + X) @ Wout^T + bout   (1536x300 @ 300x300) --------
__global__ void k8_gemm_out(const float* __restrict__ h0, const float* __restrict__ h1,
                            const float* __restrict__ X, const float* __restrict__ Wout,
                            const float* __restrict__ bout, float* __restrict__ out) {
  const int wave = (blockIdx.x * blockDim.x + threadIdx.x) >> 5;
  const int lane = threadIdx.x & 31;
  int nt = wave % 19, mt = wave / 19;
  const int l16 = lane & 15, half = lane >> 4;
  const int m0 = mt * 16, n = nt * 16 + l16;
  const bool nOK = n < Dm;
  const int nc = nOK ? n : 0;
  const int mrow = m0 + l16;
  const float* Xr = X + (size_t)mrow * Dm;
  const float* H0r = h0 + (size_t)mrow * HDm;
  const float* H1r = h1 + (size_t)mrow * HDm;
  const float* Wn = Wout + (size_t)nc * Dm;
  v8f c = {};
  for (int k0 = 0; k0 < Dm; k0 += 4) {
    const int ka = k0 + 2 * half;  // ka even; ka/ka+1 never straddle HDm
    v2f a, bb;
    const float g0 = (ka < HDm) ? H0r[ka] : H1r[ka - HDm];
    const float g1 = (ka + 1 < HDm) ? H0r[ka + 1] : H1r[ka + 1 - HDm];
    a.x = g0 + Xr[ka];
    a.y = g1 + Xr[ka + 1];
    float w0v = Wn[ka], w1v = Wn[ka + 1];
    bb.x = nOK ? w0v : 0.f;
    bb.y = nOK ? w1v : 0.f;
    c = wmma4(a, bb, c);
  }
  if (nOK) {
    const float bias = bout[n];
#pragma unroll
    for (int r = 0; r < 8; ++r) {
      const int m = m0 + r + 8 * half;
      out[(size_t)m * Dm + n] = c[r] + bias;
    }
  }
}

extern "C" void kernel_launch(void* const* d_in, const int* in_sizes, int n_in,
                              void* d_out, int out_size, void* d_ws, size_t ws_size,
                              hipStream_t stream) {
  (void)in_sizes; (void)n_in; (void)out_size; (void)ws_size;
  const float* adj  = (const float*)d_in[0];
  const float* X    = (const float*)d_in[1];
  const float* W0   = (const float*)d_in[2];
  const float* b0   = (const float*)d_in[3];
  const float* W1   = (const float*)d_in[4];
  const float* b1   = (const float*)d_in[5];
  const float* Wh   = (const float*)d_in[6];
  const float* bh   = (const float*)d_in[7];
  const float* Wout = (const float*)d_in[8];
  const float* bout = (const float*)d_in[9];
  float* out = (float*)d_out;

  float* ws  = (float*)d_ws;          // ~11.2 MB total
  float* we  = ws + 0;                // 50
  float* wh1 = ws + 64;               // 150
  float* wh2 = ws + 256;              // 150
  float* bhs = ws + 416;              // 1
  float* A20 = ws + 512;              // 589824
  float* A2w = A20 + 589824;          // 589824 (becomes A2_1 in place)
  float* Y0  = A2w + 589824;          // 460800
  float* h0  = Y0 + 460800;           // 230400
  float* s1  = h0 + 230400;           // 1536
  float* s2  = s1 + 1536;             // 1536
  float* Y1  = s2 + 1536;             // 691200
  float* h1  = Y1 + 691200;           // 230400

  k0_sums<<<1, 256, 0, stream>>>(Wh, bh, we, wh1, wh2, bhs);
  k1_adjreduce<<<2304, 256, 0, stream>>>(adj, we, A20, A2w);
  k2_gemm_y0<<<228, 256, 0, stream>>>(A20, X, Y0);        // 4*24*19 = 1824 waves
  k3_gemm_h0<<<120, 256, 0, stream>>>(Y0, W0, b0, h0);    // 96*10 = 960 waves
  k4_s12<<<6, 256, 0, stream>>>(h0, wh1, wh2, s1, s2);
  k5_a21<<<2304, 256, 0, stream>>>(A2w, s1, s2, bhs);
  k6_gemm_y1<<<348, 256, 0, stream>>>(A2w, X, h0, Y1);    // 4*24*29 = 2784 waves
  k7_gemm_h1<<<120, 256, 0, stream>>>(Y1, W1, b1, h1);    // 960 waves
  k8_gemm_out<<<228, 256, 0, stream>>>(h0, h1, X, Wout, bout, out);
}